// QuantumEntanglementLayer_90280212562550
// MI455X (gfx1250) — compile-verified
//
#include <hip/hip_runtime.h>
#include <math.h>

typedef __attribute__((ext_vector_type(2))) float v2f;
typedef __attribute__((ext_vector_type(8))) float v8f;

#define NQ   4
#define QDIM 16
#define NL   2

// ---- fast transcendental helpers (gfx1250 HW trans ops when available) ----
#if __has_builtin(__builtin_amdgcn_tanhf)
  #define FAST_TANH(x) __builtin_amdgcn_tanhf(x)
#elif __has_builtin(__builtin_amdgcn_tanh_f32)
  #define FAST_TANH(x) __builtin_amdgcn_tanh_f32(x)
#else
  #define FAST_TANH(x) tanhf(x)
#endif

// sin/cos of (2*pi*r); |r| < 0.125 here so no range reduction is needed.
#if __has_builtin(__builtin_amdgcn_sinf) && __has_builtin(__builtin_amdgcn_cosf)
  #define SIN_REV(r) __builtin_amdgcn_sinf(r)
  #define COS_REV(r) __builtin_amdgcn_cosf(r)
#else
  #define SIN_REV(r) __sinf(6.2831853071795864769f * (r))
  #define COS_REV(r) __cosf(6.2831853071795864769f * (r))
#endif

// ---------------------------------------------------------------------------
// Kernel A (one wave, runs once): fold the batch-shared part of the circuit
// (2 layers of CNOT chains + Rot gates) into a complex 16x16 matrix U.
// Thread j propagates basis column e_j; writes row-major Ure/Uim[n*16+k].
// ---------------------------------------------------------------------------
__global__ void build_unitary(const float* __restrict__ params,
                              float* __restrict__ Ure,
                              float* __restrict__ Uim) {
    const int j = threadIdx.x;
    if (j >= QDIM) return;

    float fr[QDIM], fi[QDIM];
#pragma unroll
    for (int n = 0; n < QDIM; ++n) { fr[n] = (n == j) ? 1.0f : 0.0f; fi[n] = 0.0f; }

#pragma unroll
    for (int layer = 0; layer < NL; ++layer) {
        // CNOT chain: control i -> target i+1   (new[n] = old[perm[n]])
#pragma unroll
        for (int c = 0; c < NQ - 1; ++c) {
            const int cm = 1 << (NQ - 1 - c);
            const int tm = cm >> 1;
            float tr[QDIM], ti[QDIM];
#pragma unroll
            for (int n = 0; n < QDIM; ++n) {
                const int src = (n & cm) ? (n ^ tm) : n;
                tr[n] = fr[src]; ti[n] = fi[src];
            }
#pragma unroll
            for (int n = 0; n < QDIM; ++n) { fr[n] = tr[n]; fi[n] = ti[n]; }
        }
        // Shared Rot(phi, theta, omega) = RZ(omega) RY(theta) RZ(phi) per wire
#pragma unroll
        for (int w = 0; w < NQ; ++w) {
            const float phi = params[(layer * NQ + w) * 3 + 0];
            const float th  = params[(layer * NQ + w) * 3 + 1];
            const float om  = params[(layer * NQ + w) * 3 + 2];
            float ch, sh, cap, sap, cam, sam;
            __sincosf(0.5f * th, &sh, &ch);
            __sincosf(0.5f * (phi + om), &sap, &cap);
            __sincosf(0.5f * (phi - om), &sam, &cam);
            const float g00r =  cap * ch, g00i = -sap * ch;
            const float g01r = -cam * sh, g01i = -sam * sh;
            const float g10r =  cam * sh, g10i = -sam * sh;
            const float g11r =  cap * ch, g11i =  sap * ch;
            const int tm = 1 << (NQ - 1 - w);
#pragma unroll
            for (int n0 = 0; n0 < QDIM; ++n0) {
                if (n0 & tm) continue;
                const int n1 = n0 | tm;
                const float ar = fr[n0], ai = fi[n0];
                const float br = fr[n1], bi = fi[n1];
                fr[n0] = g00r * ar - g00i * ai + g01r * br - g01i * bi;
                fi[n0] = g00r * ai + g00i * ar + g01r * bi + g01i * br;
                fr[n1] = g10r * ar - g10i * ai + g11r * br - g11i * bi;
                fi[n1] = g10r * ai + g10i * ar + g11r * bi + g11i * br;
            }
        }
        // CNOT(control 0, target 3)
        {
            const int cm = 1 << (NQ - 1);
            const int tm = 1;
            float tr[QDIM], ti[QDIM];
#pragma unroll
            for (int n = 0; n < QDIM; ++n) {
                const int src = (n & cm) ? (n ^ tm) : n;
                tr[n] = fr[src]; ti[n] = fi[src];
            }
#pragma unroll
            for (int n = 0; n < QDIM; ++n) { fr[n] = tr[n]; fi[n] = ti[n]; }
        }
    }
#pragma unroll
    for (int n = 0; n < QDIM; ++n) {
        Ure[n * QDIM + j] = fr[n];
        Uim[n * QDIM + j] = fi[n];
    }
}

// lane exchange: partner = lane ^ mask, via compile-time ds_swizzle pattern
template <int MASK>
__device__ __forceinline__ float swz_xor(float v) {
    const int p = __builtin_amdgcn_ds_swizzle(__builtin_bit_cast(int, v),
                                              (MASK << 10) | 0x1f);
    return __builtin_bit_cast(float, p);
}

// ---------------------------------------------------------------------------
// Kernel B: one wave32 per 16-sample tile.
//   A (16x4 f32 WMMA layout): lane%16 = sample row m; half h selects K pair
//     {4t+2h, 4t+2h+1}; A[m,k] = real product-state amplitude after RY gates.
//   B mirrored: lane%16 = column n; float2 load of U[n, k..k+1] (L2-resident).
//   D = A x B over 4 K-steps (fp32 WMMA), separately for Re/Im.
//   probs = Dre^2 + Dim^2, then a 16-lane Walsh butterfly per half-wave;
//   Walsh coefficients at lane indices {8,4,2,1} are <Z_0..Z_3>.
// ---------------------------------------------------------------------------
__global__ __launch_bounds__(256) void qcircuit_wmma(
    const float* __restrict__ x,
    const float* __restrict__ Ure,
    const float* __restrict__ Uim,
    float* __restrict__ out, int batch)
{
    const int lane = threadIdx.x & 31;
    const int wave = threadIdx.x >> 5;
    const int tile = blockIdx.x * 8 + wave;     // 16 samples per wave
    const int base = tile * 16;
    if (base >= batch) return;                   // uniform per wave

    const int m = lane & 15;                     // sample row / U column
    const int h = lane >> 4;                     // half-wave -> K pair select

    // ---- per-sample angles; revolutions r = tanh(x)/8  (|r| < 1/8) ----
    const float4 xv = *(const float4*)(x + (size_t)(base + m) * 4);
    const float r0 = FAST_TANH(xv.x) * 0.125f;
    const float r1 = FAST_TANH(xv.y) * 0.125f;
    const float r2 = FAST_TANH(xv.z) * 0.125f;
    const float r3 = FAST_TANH(xv.w) * 0.125f;
    const float c0 = COS_REV(r0), s0 = SIN_REV(r0);
    const float c1 = COS_REV(r1), s1 = SIN_REV(r1);
    const float c2 = COS_REV(r2), s2 = SIN_REV(r2);
    const float c3 = COS_REV(r3), s3 = SIN_REV(r3);

    // Product-state amplitudes for this lane's K set.
    // k = 4t + 2h (+0/+1): bit3=t>>1 (wire0), bit2=t&1 (wire1), bit1=h
    // (wire2), bit0 = 0/1 (wire3).
    const float f2 = h ? s2 : c2;
    const float p00 = c0 * c1 * f2, p01 = c0 * s1 * f2;
    const float p10 = s0 * c1 * f2, p11 = s0 * s1 * f2;
    v2f A[4];
    A[0] = (v2f){ p00 * c3, p00 * s3 };
    A[1] = (v2f){ p01 * c3, p01 * s3 };
    A[2] = (v2f){ p10 * c3, p10 * s3 };
    A[3] = (v2f){ p11 * c3, p11 * s3 };

    // B operands: B[k, n] = U[n, k]; contiguous float2 per step.
    v2f Br[4], Bi[4];
#pragma unroll
    for (int t = 0; t < 4; ++t) {
        const int k = 4 * t + 2 * h;
        Br[t] = *(const v2f*)(Ure + m * QDIM + k);
        Bi[t] = *(const v2f*)(Uim + m * QDIM + k);
    }

    // ---- fp32 WMMA: Re/Im amplitudes for 16 samples x 16 states ----
    v8f dre = {};
    v8f dim_ = {};
#pragma unroll
    for (int t = 0; t < 4; ++t)
        dre = __builtin_amdgcn_wmma_f32_16x16x4_f32(
            false, A[t], false, Br[t], (short)0, dre, false, false);
#pragma unroll
    for (int t = 0; t < 4; ++t)
        dim_ = __builtin_amdgcn_wmma_f32_16x16x4_f32(
            false, A[t], false, Bi[t], (short)0, dim_, false, false);

    // probs = |psi|^2 (D layout: VGPR v, half h -> sample M=v+8h; lane%16=n)
    const v8f p = dre * dre + dim_ * dim_;

    // ---- Walsh-Hadamard butterfly across 16 lanes of each half-wave ----
    float acc[8];
#pragma unroll
    for (int v = 0; v < 8; ++v) acc[v] = p[v];
#pragma unroll
    for (int bit = 0; bit < 4; ++bit) {
        const int msk = 1 << bit;
        const bool hi = (lane & msk) != 0;
#pragma unroll
        for (int v = 0; v < 8; ++v) {
            float other;
            switch (msk) {
                case 1: other = swz_xor<1>(acc[v]); break;
                case 2: other = swz_xor<2>(acc[v]); break;
                case 4: other = swz_xor<4>(acc[v]); break;
                default: other = swz_xor<8>(acc[v]); break;
            }
            const float t = hi ? -acc[v] : acc[v];
            acc[v] = other + t;
        }
    }

    // Walsh index j = 1<<(3-w) holds <Z_w>:  j=8->w0, 4->w1, 2->w2, 1->w3
    const int j = lane & 15;
    if (j == 1 || j == 2 || j == 4 || j == 8) {
        const int w = (j == 8) ? 0 : (j == 4) ? 1 : (j == 2) ? 2 : 3;
#pragma unroll
        for (int v = 0; v < 8; ++v)
            out[(size_t)(base + v + 8 * h) * 4 + w] = acc[v];
    }
}

// ---------------------------------------------------------------------------
extern "C" void kernel_launch(void* const* d_in, const int* in_sizes, int n_in,
                              void* d_out, int out_size, void* d_ws, size_t ws_size,
                              hipStream_t stream) {
    const float* x      = (const float*)d_in[0];   // [B, 4] fp32
    const float* params = (const float*)d_in[1];   // [2, 4, 3] fp32
    float* Ure = (float*)d_ws;                     // 16x16
    float* Uim = Ure + QDIM * QDIM;                // 16x16

    const int batch = in_sizes[0] / NQ;            // 1048576
    build_unitary<<<1, 32, 0, stream>>>(params, Ure, Uim);

    const int tiles  = batch / 16;                 // one wave per tile
    const int blocks = (tiles + 7) / 8;            // 8 waves per block
    qcircuit_wmma<<<blocks, 256, 0, stream>>>(x, Ure, Uim, (float*)d_out, batch);
}